// PRGC_25503515803675
// MI455X (gfx1250) — compile-verified
//
#include <hip/hip_runtime.h>

#define Bb 4
#define Ll 192
#define Hh 768
#define Rr 24
#define Mm (Bb*Ll)          // 768 rows of flattened embed

typedef __attribute__((ext_vector_type(2))) float v2f;
typedef __attribute__((ext_vector_type(8))) float v8f;

// ---------------- small prologue kernels ----------------

__global__ void k_avg(const float* __restrict__ emb, const int* __restrict__ mask,
                      float* __restrict__ avg) {
  int tid = blockIdx.x * blockDim.x + threadIdx.x;
  if (tid >= Bb * Hh) return;
  int b = tid / Hh, h = tid % Hh;
  float s = 0.f, ms = 0.f;
  for (int l = 0; l < Ll; ++l) {
    float m = (float)mask[b * Ll + l];
    s  += emb[(b * Ll + l) * Hh + h] * m;
    ms += m;
  }
  avg[tid] = s / ms;
}

__global__ void k_relh(const float* __restrict__ avg, const float* __restrict__ W,
                       const float* __restrict__ bias, float* __restrict__ out) {
  int tid = blockIdx.x * blockDim.x + threadIdx.x;
  const int N = Hh / 2;
  if (tid >= Bb * N) return;
  int b = tid / N, o = tid % N;
  float s = bias[o];
  for (int k = 0; k < Hh; ++k) s += avg[b * Hh + k] * W[k * N + o];
  out[tid] = fmaxf(s, 0.f);
}

__global__ void k_stage1(const float* __restrict__ rh, const float* __restrict__ W,
                         const float* __restrict__ bias, float* __restrict__ out) {
  int tid = blockIdx.x * blockDim.x + threadIdx.x;
  if (tid >= Bb * Rr) return;
  int b = tid / Rr, r = tid % Rr;
  float s = bias[r];
  const int K = Hh / 2;
  for (int k = 0; k < K; ++k) s += rh[b * K + k] * W[k * Rr + r];
  out[tid] = s;
}

// relc[b,h] = rel_emb[target_rel[b]] @ W_tag[:H] + b_tag   (rank-1-over-L part of fusion)
__global__ void k_relc(const int* __restrict__ trel, const float* __restrict__ rel_emb,
                       const float* __restrict__ W_tag, const float* __restrict__ b_tag,
                       float* __restrict__ relc) {
  int tid = blockIdx.x * blockDim.x + threadIdx.x;
  if (tid >= Bb * Hh) return;
  int b = tid / Hh, h = tid % Hh;
  int t = trel[b];
  float s = b_tag[h];
  for (int k = 0; k < Hh; ++k) s += rel_emb[t * Hh + k] * W_tag[k * Hh + h];
  relc[tid] = s;
}

// ---------------- fused fp32 WMMA GEMM, async double-buffered ----------------
// C[z] = X[768,768] @ W[z][768,768], z in {W_corr[:H], W_corr[H:], W_tag[H:]}
#define BM 64
#define BN 128
#define BK 32
#define AST 36      // As padded stride (floats): conflict-free A-fragment gathers
#define BST 144     // Bs padded stride (floats): half-waves hit disjoint bank groups

__global__ __launch_bounds__(256)
void k_gemm(const float* __restrict__ X,
            const float* __restrict__ w0, const float* __restrict__ w1,
            const float* __restrict__ w2,
            float* __restrict__ o0, float* __restrict__ o1, float* __restrict__ o2) {
  __shared__ float As[2][BM * AST];
  __shared__ float Bs[2][BK * BST];

  const float* W = (blockIdx.z == 0) ? w0 : (blockIdx.z == 1 ? w1 : w2);
  float*       O = (blockIdx.z == 0) ? o0 : (blockIdx.z == 1 ? o1 : o2);

  const int row0 = blockIdx.y * BM;
  const int col0 = blockIdx.x * BN;
  const int tid  = threadIdx.x;
  const int wave = tid >> 5, lane = tid & 31;
  const int wm = wave & 1;        // 2 wave-rows of 32
  const int wn = wave >> 1;       // 4 wave-cols of 32
  const int l15 = lane & 15, hi = lane >> 4;

  // LDS byte offsets of the buffers (AS3->flat lowering is {aperture_hi, offset})
  const unsigned ldsA0 = (unsigned)(uintptr_t)&As[0][0];
  const unsigned ldsA1 = (unsigned)(uintptr_t)&As[1][0];
  const unsigned ldsB0 = (unsigned)(uintptr_t)&Bs[0][0];
  const unsigned ldsB1 = (unsigned)(uintptr_t)&Bs[1][0];

  // Issue one 16-byte async global->LDS copy per lane per statement.
  auto stage = [&](int buf, int kt) {
    const unsigned la = buf ? ldsA1 : ldsA0;
    const unsigned lb = buf ? ldsB1 : ldsB0;
#pragma unroll
    for (int r = 0; r < 2; ++r) {               // A tile: 512 vec4, 2 per thread
      int v = tid + r * 256;
      int m = v >> 3, kv = (v & 7) << 2;
      unsigned loff = la + (unsigned)((m * AST + kv) * 4);
      unsigned long long g =
          (unsigned long long)(uintptr_t)(X + (row0 + m) * Hh + kt + kv);
      asm volatile("global_load_async_to_lds_b128 %0, %1, off"
                   :: "v"(loff), "v"(g) : "memory");
    }
#pragma unroll
    for (int r = 0; r < 4; ++r) {               // B tile: 1024 vec4, 4 per thread
      int v = tid + r * 256;
      int k = v >> 5, nv = (v & 31) << 2;
      unsigned loff = lb + (unsigned)((k * BST + nv) * 4);
      unsigned long long g =
          (unsigned long long)(uintptr_t)(W + (kt + k) * Hh + col0 + nv);
      asm volatile("global_load_async_to_lds_b128 %0, %1, off"
                   :: "v"(loff), "v"(g) : "memory");
    }
  };

  const v8f vz = {0.f, 0.f, 0.f, 0.f, 0.f, 0.f, 0.f, 0.f};
  v8f acc[2][2];
  acc[0][0] = vz; acc[0][1] = vz; acc[1][0] = vz; acc[1][1] = vz;

  stage(0, 0);                                   // prologue: fill buffer 0
  asm volatile("s_wait_asynccnt 0x0" ::: "memory");
  __syncthreads();

  for (int kt = 0; kt < Hh; kt += BK) {
    const int cur = (kt / BK) & 1;
    if (kt + BK < Hh) stage(cur ^ 1, kt + BK);   // prefetch next tile async

    const float* Asl = &As[cur][0];
    const float* Bsl = &Bs[cur][0];

#pragma unroll
    for (int kk = 0; kk < BK; kk += 4) {
      v2f a[2], bf[2];
#pragma unroll
      for (int i = 0; i < 2; ++i) {
        // A 16x4 layout: lanes 0-15 -> K=kk,kk+1 ; lanes 16-31 -> K=kk+2,kk+3
        int m_local = wm * 32 + i * 16 + l15;
        const float* ap = &Asl[m_local * AST + kk + hi * 2];
        a[i].x = ap[0];
        a[i].y = ap[1];
      }
#pragma unroll
      for (int j = 0; j < 2; ++j) {
        // B 4x16 layout: lane = N; v0 = K row (kk | kk+2), v1 = next K row
        int n_local = wn * 32 + j * 16 + l15;
        int bk = kk + hi * 2;
        bf[j].x = Bsl[bk * BST + n_local];
        bf[j].y = Bsl[(bk + 1) * BST + n_local];
      }
#pragma unroll
      for (int i = 0; i < 2; ++i)
#pragma unroll
        for (int j = 0; j < 2; ++j)
          acc[i][j] = __builtin_amdgcn_wmma_f32_16x16x4_f32(
              false, a[i], false, bf[j], (short)0, acc[i][j], false, false);
    }

    // Own async copies done, then block-wide handshake (also drains DScnt)
    asm volatile("s_wait_asynccnt 0x0" ::: "memory");
    __syncthreads();
  }

  // D layout: lanes 0-15 -> M=r, lanes 16-31 -> M=8+r; N = lane&15
#pragma unroll
  for (int i = 0; i < 2; ++i) {
#pragma unroll
    for (int j = 0; j < 2; ++j) {
      int rbase = row0 + wm * 32 + i * 16 + hi * 8;
      int c     = col0 + wn * 32 + j * 16 + l15;
#pragma unroll
      for (int r = 0; r < 8; ++r)
        O[(rbase + r) * Hh + c] = acc[i][j][r];
    }
  }
}

// ---------------- pred_corres: relu-inside-reduction (not a GEMM) ----------------
__global__ __launch_bounds__(256)
void k_corr(const float* __restrict__ h1, const float* __restrict__ h2,
            const float* __restrict__ b_corr, const float* __restrict__ W_gc,
            const float* __restrict__ b_gc, float* __restrict__ out) {
  __shared__ float h2s[Hh];
  __shared__ float wg[Hh];
  const int bi = blockIdx.x;          // b*L + i
  const int b  = bi / Ll;
  const int tid = threadIdx.x, wave = tid >> 5, lane = tid & 31;

  for (int h = tid; h < Hh; h += 256) {
    h2s[h] = h2[bi * Hh + h] + b_corr[h];
    wg[h]  = W_gc[h];
  }
  __syncthreads();

  const float bg = b_gc[0];
  for (int jl = 0; jl < Ll / 8; ++jl) {
    int j = jl * 8 + wave;
    const float* h1r = h1 + (b * Ll + j) * Hh;
    float s = 0.f;
    for (int h = lane; h < Hh; h += 32)
      s += fmaxf(h1r[h] + h2s[h], 0.f) * wg[h];
    for (int off = 16; off > 0; off >>= 1)
      s += __shfl_xor(s, off, 32);
    if (lane == 0) out[bi * Ll + j] = s + bg;
  }
}

// ---------------- tag heads: fusion = fe + relc[b]; 3+3-wide dots ----------------
__global__ void k_heads(const float* __restrict__ fe, const float* __restrict__ relc,
                        const float* __restrict__ W_sub, const float* __restrict__ b_sub,
                        const float* __restrict__ W_obj, const float* __restrict__ b_obj,
                        float* __restrict__ out_sub, float* __restrict__ out_obj) {
  int row = blockIdx.x * blockDim.x + threadIdx.x;
  if (row >= Mm) return;
  int b = row / Ll;
  float s0 = b_sub[0], s1 = b_sub[1], s2 = b_sub[2];
  float q0 = b_obj[0], q1 = b_obj[1], q2 = b_obj[2];
  const float* fr = fe + row * Hh;
  const float* rc = relc + b * Hh;
  for (int h = 0; h < Hh; ++h) {
    float f = fr[h] + rc[h];
    s0 += f * W_sub[h * 3 + 0];
    s1 += f * W_sub[h * 3 + 1];
    s2 += f * W_sub[h * 3 + 2];
    q0 += f * W_obj[h * 3 + 0];
    q1 += f * W_obj[h * 3 + 1];
    q2 += f * W_obj[h * 3 + 2];
  }
  out_sub[row * 3 + 0] = s0; out_sub[row * 3 + 1] = s1; out_sub[row * 3 + 2] = s2;
  out_obj[row * 3 + 0] = q0; out_obj[row * 3 + 1] = q1; out_obj[row * 3 + 2] = q2;
}

// ---------------- launcher ----------------
extern "C" void kernel_launch(void* const* d_in, const int* in_sizes, int n_in,
                              void* d_out, int out_size, void* d_ws, size_t ws_size,
                              hipStream_t stream) {
  (void)in_sizes; (void)n_in; (void)out_size; (void)ws_size;

  const float* emb    = (const float*)d_in[0];
  const int*   mask   = (const int*)  d_in[1];
  const int*   trel   = (const int*)  d_in[2];
  const float* W_relh = (const float*)d_in[3];
  const float* b_relh = (const float*)d_in[4];
  const float* W_rj   = (const float*)d_in[5];
  const float* b_rj   = (const float*)d_in[6];
  const float* W_corr = (const float*)d_in[7];
  const float* b_corr = (const float*)d_in[8];
  const float* W_gc   = (const float*)d_in[9];
  const float* b_gc   = (const float*)d_in[10];
  const float* rel_emb= (const float*)d_in[11];
  const float* W_tag  = (const float*)d_in[12];
  const float* b_tag  = (const float*)d_in[13];
  const float* W_sub  = (const float*)d_in[14];
  const float* b_sub  = (const float*)d_in[15];
  const float* W_obj  = (const float*)d_in[16];
  const float* b_obj  = (const float*)d_in[17];

  float* out    = (float*)d_out;
  float* stage1 = out;                         // [B,R] = 96
  float* subj   = out + Bb * Rr;               // [B,L,3] = 2304
  float* obj    = subj + Mm * 3;               // [B,L,3] = 2304
  float* corr   = obj + Mm * 3;                // [B,L,L,1] = 147456

  float* ws   = (float*)d_ws;
  float* h1   = ws;                 // [M,H]
  float* h2   = h1 + Mm * Hh;       // [M,H]
  float* fe   = h2 + Mm * Hh;       // [M,H]
  float* avg  = fe + Mm * Hh;       // [B,H]
  float* relh = avg + Bb * Hh;      // [B,H/2]
  float* relc = relh + Bb * (Hh/2); // [B,H]

  k_avg   <<<(Bb * Hh + 255) / 256, 256, 0, stream>>>(emb, mask, avg);
  k_relh  <<<(Bb * (Hh/2) + 255) / 256, 256, 0, stream>>>(avg, W_relh, b_relh, relh);
  k_stage1<<<1, 128, 0, stream>>>(relh, W_rj, b_rj, stage1);
  k_relc  <<<(Bb * Hh + 255) / 256, 256, 0, stream>>>(trel, rel_emb, W_tag, b_tag, relc);

  dim3 gg(Hh / BN, Mm / BM, 3);   // (6, 12, 3)
  k_gemm<<<gg, 256, 0, stream>>>(emb,
                                 W_corr,                 // W_corr[:H]
                                 W_corr + Hh * Hh,       // W_corr[H:]
                                 W_tag  + Hh * Hh,       // W_tag[H:] (embed half)
                                 h1, h2, fe);

  k_corr <<<Mm, 256, 0, stream>>>(h1, h2, b_corr, W_gc, b_gc, corr);
  k_heads<<<(Mm + 255) / 256, 256, 0, stream>>>(fe, relc, W_sub, b_sub, W_obj, b_obj,
                                                subj, obj);
}